// get_new_points_11948599018348
// MI455X (gfx1250) — compile-verified
//
#include <hip/hip_runtime.h>

#define N_PTS   65536
#define N_BATCH 16
#define M_CENT  256
#define NS      32

typedef __attribute__((ext_vector_type(2))) float v2f;
typedef __attribute__((ext_vector_type(8))) float v8f;

// ---------------------------------------------------------------------------
// 1) Furthest point sampling: one block per batch, 1024 threads.
//    Running min-distance array lives in LDS (256 KB <= 320 KB CDNA5 WGP LDS).
// ---------------------------------------------------------------------------
__global__ __launch_bounds__(1024)
void fps_kernel(const float* __restrict__ p1, int* __restrict__ fps_idx) {
  __shared__ float sdist[N_PTS];   // 256 KB
  __shared__ float swd[32];
  __shared__ int   swi[32];
  __shared__ int   s_last;
  const int b    = blockIdx.x;
  const int tid  = threadIdx.x;
  const int lane = tid & 31;
  const int wid  = tid >> 5;
  const float* pts = p1 + (size_t)b * N_PTS * 3;

  #pragma unroll
  for (int j = 0; j < N_PTS / 1024; ++j) sdist[tid + j * 1024] = 1e10f;
  if (tid == 0) fps_idx[b * M_CENT + 0] = 0;
  __syncthreads();

  int last = 0;
  for (int it = 1; it < M_CENT; ++it) {
    const float lx = pts[last * 3 + 0];
    const float ly = pts[last * 3 + 1];
    const float lz = pts[last * 3 + 2];
    float best = -1.0f;
    int   bidx = 0x7fffffff;
    #pragma unroll 4
    for (int j = 0; j < N_PTS / 1024; ++j) {
      const int   i  = tid + j * 1024;
      const float dx = pts[i * 3 + 0] - lx;
      const float dy = pts[i * 3 + 1] - ly;
      const float dz = pts[i * 3 + 2] - lz;
      const float d  = dx * dx + dy * dy + dz * dz;
      const float dm = fminf(sdist[i], d);
      sdist[i] = dm;
      if (dm > best || (dm == best && i < bidx)) { best = dm; bidx = i; }
    }
    // wave32 argmax reduction (lowest index wins ties, like jnp.argmax)
    for (int off = 16; off > 0; off >>= 1) {
      const float od = __shfl_down(best, off);
      const int   oi = __shfl_down(bidx, off);
      if (od > best || (od == best && oi < bidx)) { best = od; bidx = oi; }
    }
    if (lane == 0) { swd[wid] = best; swi[wid] = bidx; }
    __syncthreads();
    if (tid < 32) {
      float v  = swd[tid];
      int   vi = swi[tid];
      for (int off = 16; off > 0; off >>= 1) {
        const float od = __shfl_down(v, off);
        const int   oi = __shfl_down(vi, off);
        if (od > v || (od == v && oi < vi)) { v = od; vi = oi; }
      }
      if (tid == 0) { s_last = vi; fps_idx[b * M_CENT + it] = vi; }
    }
    __syncthreads();
    last = s_last;
  }
}

// ---------------------------------------------------------------------------
// 2) Gather centroids + global per-coordinate min (single block).
// ---------------------------------------------------------------------------
__global__ __launch_bounds__(256)
void gather_min_kernel(const float* __restrict__ p1, const int* __restrict__ fps_idx,
                       float* __restrict__ p2, float* __restrict__ mins) {
  __shared__ float smx[256], smy[256], smz[256];
  const int t = threadIdx.x;
  float mnx = 3.402823466e38f, mny = 3.402823466e38f, mnz = 3.402823466e38f;
  for (int r = 0; r < (N_BATCH * M_CENT) / 256; ++r) {
    const int c   = t + r * 256;
    const int b   = c / M_CENT;
    const int idx = fps_idx[c];
    const float* p = p1 + ((size_t)b * N_PTS + idx) * 3;
    const float x = p[0], y = p[1], z = p[2];
    p2[c * 3 + 0] = x; p2[c * 3 + 1] = y; p2[c * 3 + 2] = z;
    mnx = fminf(mnx, x); mny = fminf(mny, y); mnz = fminf(mnz, z);
  }
  smx[t] = mnx; smy[t] = mny; smz[t] = mnz;
  __syncthreads();
  for (int off = 128; off > 0; off >>= 1) {
    if (t < off) {
      smx[t] = fminf(smx[t], smx[t + off]);
      smy[t] = fminf(smy[t], smy[t + off]);
      smz[t] = fminf(smz[t], smz[t + off]);
    }
    __syncthreads();
  }
  if (t == 0) { mins[0] = smx[0]; mins[1] = smy[0]; mins[2] = smz[0]; }
}

// ---------------------------------------------------------------------------
// 3) Stable sort of each batch's 256 centroids by distance to global-min corner.
//    Bitonic sort on (key, idx) lexicographic pairs == stable argsort.
// ---------------------------------------------------------------------------
__global__ __launch_bounds__(256)
void sort_kernel(const float* __restrict__ p2, const float* __restrict__ mins,
                 float* __restrict__ p2s) {
  __shared__ float sk[M_CENT];
  __shared__ int   si[M_CENT];
  const int t = threadIdx.x, b = blockIdx.x;
  const float mx = mins[0], my = mins[1], mz = mins[2];
  const float* c = p2 + ((size_t)b * M_CENT + t) * 3;
  const float dx = c[0] - mx, dy = c[1] - my, dz = c[2] - mz;
  sk[t] = dx * dx + dy * dy + dz * dz;
  si[t] = t;
  __syncthreads();
  for (int k = 2; k <= M_CENT; k <<= 1) {
    for (int j = k >> 1; j > 0; j >>= 1) {
      const int ixj = t ^ j;
      if (ixj > t) {
        const float a = sk[t], bb = sk[ixj];
        const int  ai = si[t], bi = si[ixj];
        const bool agtb = (a > bb) || (a == bb && ai > bi);
        const bool up   = ((t & k) == 0);
        if (agtb == up) { sk[t] = bb; sk[ixj] = a; si[t] = bi; si[ixj] = ai; }
      }
      __syncthreads();
    }
  }
  const int src = si[t];
  const float* gsrc = p2 + ((size_t)b * M_CENT + src) * 3;
  float* o = p2s + ((size_t)b * M_CENT + t) * 3;
  o[0] = gsrc[0]; o[1] = gsrc[1]; o[2] = gsrc[2];
}

// ---------------------------------------------------------------------------
// 4) KNN + grouping. Per-wave running top-32 kept sorted across the 32 lanes;
//    candidate insertion via the shuffle min/max formula. Distances produced
//    16x16 per V_WMMA_F32_16X16X4_F32 (A = -2*centroids, C = |c|^2 + |p|^2).
// ---------------------------------------------------------------------------
__device__ __forceinline__ void topk_scan(const float* __restrict__ row, int chunk,
                                          int lane, float& rd, int& ri) {
  float thr  = __shfl(rd, 31);
  int   thri = __shfl(ri, 31);
  for (int j = 0; j < 32; ++j) {
    const float v   = row[lane + 32 * j];
    const int   idx = chunk * 1024 + lane + 32 * j;
    const bool pass = (v < thr) || (v == thr && idx < thri);
    unsigned long long mask = __ballot(pass);
    while (mask) {
      const int src = __ffsll(mask) - 1;
      mask &= mask - 1;
      const float cv   = __shfl(v, src);
      const int   cidx = __shfl(idx, src);
      // insert broadcast candidate into lane-sorted top-32:
      float pd = __shfl_up(rd, 1);
      int   pi = __shfl_up(ri, 1);
      if (lane == 0) { pd = -3.402823466e38f; pi = -1; }
      const bool cgt = (cv > pd) || (cv == pd && cidx > pi);   // lexmax(prev, cand)
      const float md = cgt ? cv : pd;
      const int   mi = cgt ? cidx : pi;
      const bool mlt = (md < rd) || (md == rd && mi < ri);     // lexmin(cur, m)
      if (mlt) { rd = md; ri = mi; }
      thr  = __shfl(rd, 31);
      thri = __shfl(ri, 31);
    }
  }
}

__global__ __launch_bounds__(256)
void knn_group_kernel(const float* __restrict__ p1, const float* __restrict__ p2s,
                      float* __restrict__ out) {
  __shared__ float s_d2[16 * 1024];  // 64 KB: 16 centroids x 1024-point chunk
  __shared__ float s_cn[16];
  const int g = blockIdx.x, b = blockIdx.y;
  const int tid  = threadIdx.x;
  const int lane = tid & 31, wave = tid >> 5;
  const int ci = lane & 15, half = lane >> 4;
  const float* pts  = p1  + (size_t)b * N_PTS * 3;
  const float* cent = p2s + ((size_t)b * M_CENT + g * 16) * 3;

  if (tid < 16) {
    const float x = cent[tid * 3 + 0], y = cent[tid * 3 + 1], z = cent[tid * 3 + 2];
    s_cn[tid] = x * x + y * y + z * z;
  }
  __syncthreads();

  // A operand: 16x4 f32, lane holds centroid M=lane%16 at K=2*half+j; pad K=3 with 0.
  const float cx = cent[ci * 3 + 0], cy = cent[ci * 3 + 1], cz = cent[ci * 3 + 2];
  v2f A;
  A.x = -2.0f * (half ? cz : cx);
  A.y = half ? 0.0f : (-2.0f * cy);
  float cn[8];
  #pragma unroll
  for (int v = 0; v < 8; ++v) cn[v] = s_cn[v + 8 * half];  // C layout: M = v + 8*(lane>=16)

  float rd0 = 3.402823466e38f, rd1 = 3.402823466e38f;
  int   ri0 = 0x7fffffff,      ri1 = 0x7fffffff;
  const int c0 = 2 * wave, c1 = 2 * wave + 1;

  for (int chunk = 0; chunk < N_PTS / 1024; ++chunk) {
    #pragma unroll
    for (int s = 0; s < 8; ++s) {
      const int lt = wave * 8 + s;                 // tile within chunk
      const int pi = chunk * 1024 + lt * 16 + ci;  // point index, N = lane%16
      const float px = pts[pi * 3 + 0], py = pts[pi * 3 + 1], pz = pts[pi * 3 + 2];
      v2f B;                                       // 4x16 f32, same K mapping as A
      B.x = half ? pz : px;
      B.y = half ? 0.0f : py;
      const float pn = px * px + py * py + pz * pz;
      v8f C;
      #pragma unroll
      for (int v = 0; v < 8; ++v) C[v] = cn[v] + pn;
      // D = |c|^2 + |p|^2 - 2 c.p  for a 16x16 (centroid x point) tile
      v8f D = __builtin_amdgcn_wmma_f32_16x16x4_f32(
          false, A, false, B, (short)0, C, false, false);
      #pragma unroll
      for (int v = 0; v < 8; ++v)
        s_d2[(v + 8 * half) * 1024 + lt * 16 + ci] = D[v];
    }
    __syncthreads();
    topk_scan(&s_d2[c0 * 1024], chunk, lane, rd0, ri0);
    topk_scan(&s_d2[c1 * 1024], chunk, lane, rd1, ri1);
    __syncthreads();
  }

  // Emit: out[b][m][coord*32 + k] = p1[knn_k] - centroid  (lane k = k-th nearest)
  #pragma unroll
  for (int which = 0; which < 2; ++which) {
    const int c  = 2 * wave + which;
    const int ri = which ? ri1 : ri0;
    const float ccx = cent[c * 3 + 0], ccy = cent[c * 3 + 1], ccz = cent[c * 3 + 2];
    const float* np = pts + (size_t)ri * 3;
    const int m = g * 16 + c;
    const size_t base = (size_t)(b * M_CENT + m) * 3 * NS;
    out[base + 0 * NS + lane] = np[0] - ccx;
    out[base + 1 * NS + lane] = np[1] - ccy;
    out[base + 2 * NS + lane] = np[2] - ccz;
  }
}

// ---------------------------------------------------------------------------
extern "C" void kernel_launch(void* const* d_in, const int* in_sizes, int n_in,
                              void* d_out, int out_size, void* d_ws, size_t ws_size,
                              hipStream_t stream) {
  (void)in_sizes; (void)n_in; (void)out_size; (void)ws_size;
  const float* p1 = (const float*)d_in[0];
  float* out = (float*)d_out;
  char* ws = (char*)d_ws;

  int*   fps_idx = (int*)(ws);                     // 4096 ints   (16 KB)
  float* p2      = (float*)(ws + 16384);           // 12288 f32   (48 KB)
  float* mins    = (float*)(ws + 16384 + 49152);   // 4 f32
  float* p2s     = (float*)(ws + 16384 + 49152 + 64);  // 12288 f32 (48 KB)

  fps_kernel<<<N_BATCH, 1024, 0, stream>>>(p1, fps_idx);
  gather_min_kernel<<<1, 256, 0, stream>>>(p1, fps_idx, p2, mins);
  sort_kernel<<<N_BATCH, 256, 0, stream>>>(p2, mins, p2s);
  knn_group_kernel<<<dim3(M_CENT / 16, N_BATCH), 256, 0, stream>>>(p1, p2s, out);
}